// GIN_43834436223109
// MI455X (gfx1250) — compile-verified
//
#include <hip/hip_runtime.h>
#include <math.h>

#define HD 128      // hidden / feature dim
#define GG 256      // number of graphs (pool groups)
#define CC 10       // classes
#define PSTRIDE 144 // LDS pair-row stride in float2s: +1 pair-row => +32 banks -> conflict-free

typedef float v2f __attribute__((ext_vector_type(2)));
typedef float v8f __attribute__((ext_vector_type(8)));

// ---------------- elementwise / utility kernels ----------------

__global__ void zero_f32(float* __restrict__ p, long long n) {
    long long i = blockIdx.x * (long long)blockDim.x + threadIdx.x;
    long long stride = (long long)gridDim.x * blockDim.x;
    for (; i < n; i += stride) p[i] = 0.0f;
}

__global__ void init_pre(const float* __restrict__ x, const float* __restrict__ eps,
                         int layer, float* __restrict__ pre, long long n) {
    float s = 1.0f + eps[layer];
    long long i = blockIdx.x * (long long)blockDim.x + threadIdx.x;
    long long stride = (long long)gridDim.x * blockDim.x;
    for (; i < n; i += stride) pre[i] = s * x[i];
}

// one edge handled by 32 consecutive threads, 4 floats each (float4 gather row).
// Software-pipelined prefetch (global_prefetch_b8) of the source row PF edges ahead
// hides the data-dependent gather latency; rows are L2-resident (25.6 MB << 192 MB).
__global__ void scatter_add(const float* __restrict__ x, const int* __restrict__ ei,
                            float* __restrict__ pre, int nE) {
    long long tid = blockIdx.x * (long long)blockDim.x + threadIdx.x;
    long long total = (long long)nE * 32;
    if (tid >= total) return;
    int e = (int)(tid >> 5);
    int lane = (int)(tid & 31);

    const int PF = 4096;
    int e2 = e + PF;
    if (e2 < nE) {
        int s2 = ei[e2];
        __builtin_prefetch(x + (long long)s2 * HD + lane * 4, 0, 3);
    }

    int s = ei[e];          // src
    int d = ei[nE + e];     // dst
    const float4 v = *(const float4*)(x + (long long)s * HD + lane * 4);
    float* o = pre + (long long)d * HD + lane * 4;
    atomicAdd(o + 0, v.x);
    atomicAdd(o + 1, v.y);
    atomicAdd(o + 2, v.z);
    atomicAdd(o + 3, v.w);
}

// ---------------- fp32 WMMA GEMM ----------------
// out[M x 128] = relu(A[M x 128] @ W[128 x 128] + bias)
// 256-thread block = 8 waves. W staged once into LDS as K-pair-packed float2:
//   Wl[pr*PSTRIDE + n] = (W[2pr][n], W[2pr+1][n])
// so each B operand is one aligned ds_load_b64 landing directly as an even-aligned
// v2f (no operand-assembly moves), and each A operand is one global_load_b64.
// Half-waves read pair-rows pr and pr+1: +PSTRIDE float2 = +288 dwords = +32 banks
// -> the two halves cover complementary 32-bank spans, conflict-free.
// Safe for out == A: each wave only reads/writes its own 16 rows; the only
// barrier is after W staging.

__global__ void __launch_bounds__(256)
gemm16_wmma(const float* __restrict__ A, const float* __restrict__ W,
            const float* __restrict__ bias, float* __restrict__ out,
            int nStrips, int do_relu) {
    __shared__ v2f Wl[(HD / 2) * PSTRIDE];

    // ---- stage W: transpose K-pairs while storing; coalesced float4 global reads ----
    {
        const int tid = threadIdx.x;
        for (int idx = tid; idx < (HD / 2) * (HD / 4); idx += 256) {
            int pr = idx >> 5;           // pair-row 0..63
            int q  = idx & 31;           // float4 column group
            const float4 r0 = *(const float4*)(W + (2 * pr) * HD + q * 4);
            const float4 r1 = *(const float4*)(W + (2 * pr + 1) * HD + q * 4);
            int base = pr * PSTRIDE + q * 4;
            v2f t;
            t.x = r0.x; t.y = r1.x; Wl[base + 0] = t;
            t.x = r0.y; t.y = r1.y; Wl[base + 1] = t;
            t.x = r0.z; t.y = r1.z; Wl[base + 2] = t;
            t.x = r0.w; t.y = r1.w; Wl[base + 3] = t;
        }
    }
    __syncthreads();

    const int strip = blockIdx.x * 8 + (threadIdx.x >> 5);   // wave-uniform
    if (strip >= nStrips) return;

    const int m0 = strip * 16;
    const int lane = threadIdx.x & 31;
    const int mr = lane & 15;       // row within A-tile / col within B/C tile
    const int kh = lane >> 4;       // K half: lanes 0-15 -> K=0,1 ; lanes 16-31 -> K=2,3

    v8f acc[8] = {};

    const float* arow = A + (long long)(m0 + mr) * HD;
#pragma unroll 4
    for (int k = 0; k < HD; k += 4) {
        const int ka = k + kh * 2;          // even
        const int pr = (k >> 1) + kh;       // LDS pair-row
        v2f a = *(const v2f*)(arow + ka);   // one global_load_b64
#pragma unroll
        for (int t = 0; t < 8; ++t) {
            v2f b = Wl[pr * PSTRIDE + t * 16 + mr];   // one ds_load_b64
            acc[t] = __builtin_amdgcn_wmma_f32_16x16x4_f32(
                false, a, false, b, (short)0, acc[t], false, false);
        }
    }

    // C/D layout: VGPR r holds M = r (lanes 0-15) or M = 8+r (lanes 16-31); N = lane%16
    const int mbase = m0 + kh * 8;
#pragma unroll
    for (int t = 0; t < 8; ++t) {
        const int n = t * 16 + mr;
        const float bv = bias[n];
#pragma unroll
        for (int r = 0; r < 8; ++r) {
            float v = acc[t][r] + bv;
            if (do_relu) v = fmaxf(v, 0.0f);
            out[(long long)(mbase + r) * HD + n] = v;
        }
    }
}

// ---------------- BatchNorm (biased variance, torch-style) ----------------

__global__ void bn_stats(const float* __restrict__ h, float* __restrict__ st, int N) {
    int c = threadIdx.x;  // blockDim.x == 128
    float s = 0.0f, sq = 0.0f;
    for (int r = blockIdx.x; r < N; r += gridDim.x) {
        float v = h[(long long)r * HD + c];
        s += v;
        sq += v * v;
    }
    atomicAdd(&st[c], s);
    atomicAdd(&st[HD + c], sq);
}

__global__ void bn_apply(float* __restrict__ h, const float* __restrict__ st,
                         const float* __restrict__ g, const float* __restrict__ bt,
                         float invN, long long n) {
    long long i = blockIdx.x * (long long)blockDim.x + threadIdx.x;
    long long stride = (long long)gridDim.x * blockDim.x;
    for (; i < n; i += stride) {
        int c = (int)(i & (HD - 1));
        float mu = st[c] * invN;
        float var = st[HD + c] * invN - mu * mu;
        float sc = g[c] * rsqrtf(var + 1e-5f);
        h[i] = (h[i] - mu) * sc + bt[c];
    }
}

// ---------------- global mean pool ----------------

__global__ void pool_scatter(const float* __restrict__ h, const int* __restrict__ batch,
                             float* __restrict__ psum, float* __restrict__ pcnt, int N) {
    long long i = blockIdx.x * (long long)blockDim.x + threadIdx.x;
    long long n = (long long)N * HD;
    if (i >= n) return;
    int node = (int)(i >> 7);
    int c = (int)(i & (HD - 1));
    int b = batch[node];
    atomicAdd(&psum[(long long)b * HD + c], h[i]);
    if (c == 0) atomicAdd(&pcnt[b], 1.0f);
}

__global__ void pool_div(const float* __restrict__ psum, const float* __restrict__ pcnt,
                         float* __restrict__ pooled) {
    int i = blockIdx.x * blockDim.x + threadIdx.x;
    if (i >= GG * HD) return;
    int g = i >> 7;
    pooled[i] = psum[i] / fmaxf(pcnt[g], 1.0f);
}

// ---------------- final 128->10 projection + log_softmax ----------------

__global__ void head(const float* __restrict__ z1, const float* __restrict__ w,
                     const float* __restrict__ bias, float* __restrict__ out) {
    int g = blockIdx.x * blockDim.x + threadIdx.x;
    if (g >= GG) return;
    const float* row = z1 + (long long)g * HD;
    float z[CC];
#pragma unroll
    for (int c = 0; c < CC; ++c) z[c] = bias[c];
    for (int k = 0; k < HD; ++k) {
        float a = row[k];
#pragma unroll
        for (int c = 0; c < CC; ++c) z[c] += a * w[k * CC + c];
    }
    float mx = z[0];
#pragma unroll
    for (int c = 1; c < CC; ++c) mx = fmaxf(mx, z[c]);
    float se = 0.0f;
#pragma unroll
    for (int c = 0; c < CC; ++c) se += expf(z[c] - mx);
    float ls = mx + logf(se);
#pragma unroll
    for (int c = 0; c < CC; ++c) out[g * CC + c] = z[c] - ls;
}

// ---------------- driver ----------------

extern "C" void kernel_launch(void* const* d_in, const int* in_sizes, int n_in,
                              void* d_out, int out_size, void* d_ws, size_t ws_size,
                              hipStream_t stream) {
    const float* x    = (const float*)d_in[0];
    const int*  ei    = (const int*)d_in[1];     // [2, E] int32 (jax x64 disabled)
    const int*  batch = (const int*)d_in[2];
    const float* eps  = (const float*)d_in[21];
    const float* lw1  = (const float*)d_in[22];
    const float* lb1  = (const float*)d_in[23];
    const float* lw2  = (const float*)d_in[24];
    const float* lb2  = (const float*)d_in[25];

    const int N  = in_sizes[0] / HD;
    const int nE = in_sizes[1] / 2;
    const long long NH = (long long)N * HD;

    // workspace layout (floats)
    float* bufA   = (float*)d_ws;                   // N*H
    float* bufB   = bufA + NH;                      // N*H
    float* st     = bufB + NH;                      // 2*H   (sum, sumsq)
    float* psum   = st + 2 * HD;                    // G*H
    float* pcnt   = psum + (long long)GG * HD;      // G
    float* pooled = pcnt + GG;                      // G*H
    float* z1     = pooled + (long long)GG * HD;    // G*H

    const int elemGrid = 2048;
    const float invN = 1.0f / (float)N;

    const int nStrips = N / 16;                 // 3125 (M multiple of 16)
    const int gemmBlocks = (nStrips + 7) / 8;   // 8 waves per block

    const float* lin = x;
    float* cur = nullptr;
    for (int l = 0; l < 3; ++l) {
        const float* wa = (const float*)d_in[3 + 6 * l];
        const float* ba = (const float*)d_in[4 + 6 * l];
        const float* wb = (const float*)d_in[5 + 6 * l];
        const float* bb = (const float*)d_in[6 + 6 * l];
        const float* gg = (const float*)d_in[7 + 6 * l];
        const float* bt = (const float*)d_in[8 + 6 * l];
        cur = (l == 1) ? bufB : bufA;   // l0->A, l1->B, l2->A (ping-pong, in != out)

        init_pre<<<elemGrid, 256, 0, stream>>>(lin, eps, l, cur, NH);

        long long sthreads = (long long)nE * 32;
        int sblocks = (int)((sthreads + 255) / 256);
        scatter_add<<<sblocks, 256, 0, stream>>>(lin, ei, cur, nE);

        gemm16_wmma<<<gemmBlocks, 256, 0, stream>>>(cur, wa, ba, cur, nStrips, 1);
        gemm16_wmma<<<gemmBlocks, 256, 0, stream>>>(cur, wb, bb, cur, nStrips, 1);

        zero_f32<<<1, 256, 0, stream>>>(st, 2 * HD);
        bn_stats<<<512, HD, 0, stream>>>(cur, st, N);
        bn_apply<<<elemGrid, 256, 0, stream>>>(cur, st, gg, bt, invN, NH);

        lin = cur;
    }

    // global mean pool
    zero_f32<<<64, 256, 0, stream>>>(psum, (long long)GG * HD + GG);
    pool_scatter<<<(int)((NH + 255) / 256), 256, 0, stream>>>(lin, batch, psum, pcnt, N);
    pool_div<<<(GG * HD + 255) / 256, 256, 0, stream>>>(psum, pcnt, pooled);

    // head: relu(pooled @ lw1 + lb1) via WMMA (16 strips -> 2 blocks), then 128->10 + log_softmax
    gemm16_wmma<<<(GG / 16 + 7) / 8, 256, 0, stream>>>(pooled, lw1, lb1, z1, GG / 16, 1);
    head<<<(GG + 63) / 64, 64, 0, stream>>>(z1, lw2, lb2, (float*)d_out);
}